// AutoEncoder_65094524338526
// MI455X (gfx1250) — compile-verified
//
#include <hip/hip_runtime.h>
#include <math.h>

typedef __attribute__((ext_vector_type(2))) float v2f;
typedef __attribute__((ext_vector_type(8))) float v8f;

static constexpr int Hc = 128, Wc = 128;

__device__ __forceinline__ v8f wmma4(v2f a, v2f b, v8f c) {
  // D = A(16x4 f32) * B(4x16 f32) + C(16x16 f32)
  return __builtin_amdgcn_wmma_f32_16x16x4_f32(false, a, false, b, (short)0, c,
                                               false, false);
}

__device__ __forceinline__ float sigm(float x) {
  return 1.0f / (1.0f + __expf(-x));
}

// ---------------------------------------------------------------------------
// Weight repack: HWIO [3,3,CIN,COUT] -> WMMA B-fragment order.
// pk[((kk*(CIN/4)+kc4)*(COUT/16)+ct)*32 + lane] (v2f) =
//   { w[kk][kc4*4 + 2*(lane>>4)][ct*16 + (lane&15)],
//     w[kk][kc4*4 + 2*(lane>>4) + 1][ct*16 + (lane&15)] }
// => each lane's B fragment is ONE coalesced global_load_b64.
// ---------------------------------------------------------------------------
template <int CIN, int COUT>
__global__ __launch_bounds__(256) void pack_w(const float* __restrict__ w,
                                              float* __restrict__ pk) {
  constexpr int NT = COUT / 16;
  constexpr int NC = CIN / 4;
  const int total = 9 * NC * NT * 32;
  int idx = blockIdx.x * 256 + threadIdx.x;
  if (idx >= total) return;
  const int lane = idx & 31;
  int rest = idx >> 5;
  const int ct = rest % NT; rest /= NT;
  const int kc4 = rest % NC; rest /= NC;
  const int kk = rest;  // 0..8 (ky*3+kx)
  const int co = ct * 16 + (lane & 15);
  const int k0 = kc4 * 4 + 2 * (lane >> 4);
  const float* src = w + (size_t)kk * CIN * COUT;
  v2f v;
  v.x = src[(size_t)k0 * COUT + co];
  v.y = src[(size_t)(k0 + 1) * COUT + co];
  *(v2f*)(pk + 2 * (size_t)idx) = v;
}

// ---------------------------------------------------------------------------
// Fused 3x3 SAME conv + bias + LeakyReLU(0.2) + BatchNorm.
// One wave computes 2 x 16 pixels x COUT channels; B fragments are loaded
// once per k-chunk and reused by both M-tiles.
// grid = N * H * (W/32), block = 32.
// ---------------------------------------------------------------------------
template <int CIN, int CSTR, int COUT>
__global__ __launch_bounds__(32) void conv3x3_bn(
    const float* __restrict__ in, const float* __restrict__ pkw,
    const float* __restrict__ bias, const float* __restrict__ gam,
    const float* __restrict__ bet, const float* __restrict__ mu,
    const float* __restrict__ var, float* __restrict__ out) {
  constexpr int NT = COUT / 16;
  constexpr int NC = CIN / 4;
  const int lane = threadIdx.x;
  const int nlo = lane & 15;
  const int hi = lane >> 4;
  constexpr int TW = Wc / 32;
  int tid = blockIdx.x;
  const int w0 = (tid % TW) * 32; tid /= TW;
  const int h  = tid % Hc;        tid /= Hc;
  const int n  = tid;

  v8f acc[2][NT];
  const v8f zz = {0.f, 0.f, 0.f, 0.f, 0.f, 0.f, 0.f, 0.f};
#pragma unroll
  for (int mt = 0; mt < 2; ++mt)
#pragma unroll
    for (int ct = 0; ct < NT; ++ct) acc[mt][ct] = zz;

#pragma unroll
  for (int ky = 0; ky < 3; ++ky) {
    const int ih = h + ky - 1;
    const int ihc = min(max(ih, 0), Hc - 1);
    const bool okh = (unsigned)ih < (unsigned)Hc;
    const float* rowBase = in + (size_t)(n * Hc + ihc) * Wc * CSTR;
    if (ky < 2)  // warm next input row into the caches
      __builtin_prefetch(in + (size_t)(n * Hc + min(ih + 1, Hc - 1)) * Wc * CSTR +
                             (size_t)w0 * CSTR, 0, 0);
#pragma unroll
    for (int kx = 0; kx < 3; ++kx) {
      const int iw0 = w0 + nlo + kx - 1;
      const int iw1 = iw0 + 16;
      const float okf0 =
          (okh && ((unsigned)iw0 < (unsigned)Wc)) ? 1.f : 0.f;
      const float okf1 =
          (okh && ((unsigned)iw1 < (unsigned)Wc)) ? 1.f : 0.f;
      const float* ap0 = rowBase + (size_t)min(max(iw0, 0), Wc - 1) * CSTR + 2 * hi;
      const float* ap1 = rowBase + (size_t)min(max(iw1, 0), Wc - 1) * CSTR + 2 * hi;
      const v2f* bp = (const v2f*)pkw +
                      (size_t)(ky * 3 + kx) * NC * NT * 32 + lane;
#pragma unroll
      for (int kc4 = 0; kc4 < NC; ++kc4) {
        v2f a0 = *(const v2f*)(ap0 + kc4 * 4);
        v2f a1 = *(const v2f*)(ap1 + kc4 * 4);
        a0.x *= okf0; a0.y *= okf0;
        a1.x *= okf1; a1.y *= okf1;
#pragma unroll
        for (int ct = 0; ct < NT; ++ct) {
          v2f b = bp[(size_t)(kc4 * NT + ct) * 32];  // shared by both M-tiles
          acc[0][ct] = wmma4(a0, b, acc[0][ct]);
          acc[1][ct] = wmma4(a1, b, acc[1][ct]);
        }
      }
    }
  }

#pragma unroll
  for (int ct = 0; ct < NT; ++ct) {
    const int ch = ct * 16 + nlo;
    const float bi = bias[ch];
    const float sc = gam[ch] * rsqrtf(var[ch] + 1e-3f);
    const float sh = bet[ch] - mu[ch] * sc;
#pragma unroll
    for (int mt = 0; mt < 2; ++mt) {
#pragma unroll
      for (int r = 0; r < 8; ++r) {
        const int m = mt * 16 + r + 8 * hi;  // C/D layout: vgpr r -> rows r,r+8
        float v = acc[mt][ct][r] + bi;
        v = (v >= 0.f) ? v : 0.2f * v;
        out[(size_t)((n * Hc + h) * Wc + w0 + m) * COUT + ch] = v * sc + sh;
      }
    }
  }
}

// ---------------------------------------------------------------------------
// Fused ConvLSTM2D timestep (filters = 32, gates = 128).
// One wave owns all 128 gate channels for 2 x 16 pixels:
//   g = conv3x3(x_t, wx) + conv3x3(h_{t-1}, wh) + b
//   i,f,o = sigmoid; cand = tanh; c = f*c + i*cand; h = o*tanh(c)
// grid = B * H * (W/32), block = 32.
// ---------------------------------------------------------------------------
template <int CINX>
__global__ __launch_bounds__(32) void lstm_step(
    const float* __restrict__ x, int xBStr,   // x_t base, batch stride
    const float* __restrict__ pkwx,           // packed [3,3,CINX,128]
    const float* __restrict__ hp, int hBStr,  // h_{t-1} base (slice t-1)
    const float* __restrict__ pkwh,           // packed [3,3,32,128]
    const float* __restrict__ bias,           // [128]
    float* __restrict__ cbuf,                 // [B,H,W,32] cell state
    float* __restrict__ hout,                 // h_t base (slice t)
    int first) {
  constexpr int NCX = CINX / 4;
  const int lane = threadIdx.x;
  const int nlo = lane & 15;
  const int hi = lane >> 4;
  constexpr int TW = Wc / 32;
  int tid = blockIdx.x;
  const int w0 = (tid % TW) * 32; tid /= TW;
  const int h  = tid % Hc;        tid /= Hc;
  const int n  = tid;

  v8f acc[2][8];
  const v8f zz = {0.f, 0.f, 0.f, 0.f, 0.f, 0.f, 0.f, 0.f};
#pragma unroll
  for (int mt = 0; mt < 2; ++mt)
#pragma unroll
    for (int j = 0; j < 8; ++j) acc[mt][j] = zz;

  // ---- input-to-gate conv (K = 9*CINX), then recurrent conv (K = 9*32) ----
#pragma unroll
  for (int pass = 0; pass < 2; ++pass) {
    if (pass == 1 && first) break;  // h_{t-1} == 0 at t==0 (uniform branch)
    const float* src = (pass == 0) ? x : hp;
    const int bstr = (pass == 0) ? xBStr : hBStr;
    const int cstr = (pass == 0) ? CINX : 32;
    const int nc = (pass == 0) ? NCX : 8;
    const float* pkw = (pass == 0) ? pkwx : pkwh;
#pragma unroll
    for (int ky = 0; ky < 3; ++ky) {
      const int ih = h + ky - 1;
      const int ihc = min(max(ih, 0), Hc - 1);
      const bool okh = (unsigned)ih < (unsigned)Hc;
      const float* rowBase = src + (size_t)n * bstr + (size_t)ihc * Wc * cstr;
      if (ky < 2)
        __builtin_prefetch(src + (size_t)n * bstr +
                               (size_t)min(ih + 1, Hc - 1) * Wc * cstr +
                               (size_t)w0 * cstr, 0, 0);
#pragma unroll
      for (int kx = 0; kx < 3; ++kx) {
        const int iw0 = w0 + nlo + kx - 1;
        const int iw1 = iw0 + 16;
        const float okf0 =
            (okh && ((unsigned)iw0 < (unsigned)Wc)) ? 1.f : 0.f;
        const float okf1 =
            (okh && ((unsigned)iw1 < (unsigned)Wc)) ? 1.f : 0.f;
        const float* ap0 = rowBase + (size_t)min(max(iw0, 0), Wc - 1) * cstr + 2 * hi;
        const float* ap1 = rowBase + (size_t)min(max(iw1, 0), Wc - 1) * cstr + 2 * hi;
        const v2f* bp = (const v2f*)pkw +
                        (size_t)(ky * 3 + kx) * nc * 8 * 32 + lane;
#pragma unroll 2
        for (int kc4 = 0; kc4 < nc; ++kc4) {
          v2f a0 = *(const v2f*)(ap0 + kc4 * 4);
          v2f a1 = *(const v2f*)(ap1 + kc4 * 4);
          a0.x *= okf0; a0.y *= okf0;
          a1.x *= okf1; a1.y *= okf1;
#pragma unroll
          for (int j = 0; j < 8; ++j) {
            v2f b = bp[(size_t)(kc4 * 8 + j) * 32];  // shared by both M-tiles
            acc[0][j] = wmma4(a0, b, acc[0][j]);
            acc[1][j] = wmma4(a1, b, acc[1][j]);
          }
        }
      }
    }
  }

  // ---- gate nonlinearities + state update ----
  // acc[*][0..1]=i, [2..3]=f, [4..5]=cand, [6..7]=o
  const size_t pixBase = (size_t)(h * Wc + w0);
#pragma unroll
  for (int mt = 0; mt < 2; ++mt) {
#pragma unroll
    for (int r = 0; r < 8; ++r) {
      const int m = mt * 16 + r + 8 * hi;
      const size_t pix = pixBase + m;
#pragma unroll
      for (int j = 0; j < 2; ++j) {
        const int ch = 16 * j + nlo;  // 0..31 within each gate
        const float gi = sigm(acc[mt][j][r] + bias[ch]);
        const float gf = sigm(acc[mt][j + 2][r] + bias[32 + ch]);
        const float gc = tanhf(acc[mt][j + 4][r] + bias[64 + ch]);
        const float go = sigm(acc[mt][j + 6][r] + bias[96 + ch]);
        const size_t cidx = ((size_t)n * (Hc * Wc) + pix) * 32 + ch;
        const float cp = first ? 0.f : cbuf[cidx];
        const float cn = gf * cp + gi * gc;
        cbuf[cidx] = cn;
        hout[(size_t)n * hBStr + pix * 32 + ch] = go * tanhf(cn);
      }
    }
  }
}

extern "C" void kernel_launch(void* const* d_in, const int* in_sizes, int n_in,
                              void* d_out, int out_size, void* d_ws,
                              size_t ws_size, hipStream_t stream) {
  (void)in_sizes; (void)n_in; (void)out_size; (void)ws_size;
  const float* x    = (const float*)d_in[0];
  const float* c1w  = (const float*)d_in[1];
  const float* c1b  = (const float*)d_in[2];
  const float* bn1g = (const float*)d_in[3];
  const float* bn1b = (const float*)d_in[4];
  const float* bn1m = (const float*)d_in[5];
  const float* bn1v = (const float*)d_in[6];
  const float* l1wx = (const float*)d_in[7];
  const float* l1wh = (const float*)d_in[8];
  const float* l1b  = (const float*)d_in[9];
  const float* c2w  = (const float*)d_in[10];
  const float* c2b  = (const float*)d_in[11];
  const float* bn2g = (const float*)d_in[12];
  const float* bn2b = (const float*)d_in[13];
  const float* bn2m = (const float*)d_in[14];
  const float* bn2v = (const float*)d_in[15];
  const float* l2wx = (const float*)d_in[16];
  const float* l2wh = (const float*)d_in[17];
  const float* l2b  = (const float*)d_in[18];
  const float* decw = (const float*)d_in[19];
  const float* decb = (const float*)d_in[20];
  const float* bn3g = (const float*)d_in[21];
  const float* bn3b = (const float*)d_in[22];
  const float* bn3m = (const float*)d_in[23];
  const float* bn3v = (const float*)d_in[24];

  const int B = 4, T = 8;
  const size_t HW = (size_t)Hc * Wc;
  const int N = B * T;  // 32 images

  float* ws   = (float*)d_ws;
  float* h1   = ws;                          // [32,H,W,48]
  float* l1h  = h1  + (size_t)N * HW * 48;   // [B,T,H,W,32]
  float* enc  = l1h + (size_t)N * HW * 32;   // [32,H,W,16]
  float* cbuf = enc + (size_t)N * HW * 16;   // [B,H,W,32]
  float* pkc1 = cbuf + (size_t)B * HW * 32;  // packed weights (~0.8 MB total)
  float* pkc2 = pkc1 + 9 * 64 * 48;
  float* pkdec= pkc2 + 9 * 32 * 16;
  float* pkx1 = pkdec+ 9 * 16 * 64;
  float* pkh1 = pkx1 + 9 * 48 * 128;
  float* pkx2 = pkh1 + 9 * 32 * 128;
  float* pkh2 = pkx2 + 9 * 16 * 128;
  float* dbuf = ws;  // reuse h1 region: [B,T,H,W,32] (h1 dead by then)

  // ---- repack all weights into WMMA B-fragment order (one b64 per frag) ----
  {
    auto g = [](int tot) { return (tot + 255) / 256; };
    pack_w<64, 48><<<g(9*16*3*32), 256, 0, stream>>>(c1w, pkc1);
    pack_w<32, 16><<<g(9*8*1*32), 256, 0, stream>>>(c2w, pkc2);
    pack_w<16, 64><<<g(9*4*4*32), 256, 0, stream>>>(decw, pkdec);
    pack_w<48, 128><<<g(9*12*8*32), 256, 0, stream>>>(l1wx, pkx1);
    pack_w<32, 128><<<g(9*8*8*32), 256, 0, stream>>>(l1wh, pkh1);
    pack_w<16, 128><<<g(9*4*8*32), 256, 0, stream>>>(l2wx, pkx2);
    pack_w<32, 128><<<g(9*8*8*32), 256, 0, stream>>>(l2wh, pkh2);
  }

  dim3 blk(32);
  const int convGrid = N * Hc * (Wc / 32);
  const int lstmGrid = B * Hc * (Wc / 32);

  // encoder conv1: 64 -> 48, bias+lrelu+bn1
  conv3x3_bn<64, 64, 48><<<convGrid, blk, 0, stream>>>(
      x, pkc1, c1b, bn1g, bn1b, bn1m, bn1v, h1);

  // ConvLSTM 1: input h1 (48 ch), output l1h (32 ch)
  for (int t = 0; t < T; ++t) {
    lstm_step<48><<<lstmGrid, blk, 0, stream>>>(
        h1 + (size_t)t * HW * 48, (int)(T * HW * 48), pkx1,
        (t == 0) ? l1h : l1h + (size_t)(t - 1) * HW * 32, (int)(T * HW * 32),
        pkh1, l1b, cbuf, l1h + (size_t)t * HW * 32, (t == 0) ? 1 : 0);
  }

  // encoder conv2: 32 -> 16, bias+lrelu+bn2
  conv3x3_bn<32, 32, 16><<<convGrid, blk, 0, stream>>>(
      l1h, pkc2, c2b, bn2g, bn2b, bn2m, bn2v, enc);

  // ConvLSTM 2: input enc (16 ch), output dbuf (32 ch)
  for (int t = 0; t < T; ++t) {
    lstm_step<16><<<lstmGrid, blk, 0, stream>>>(
        enc + (size_t)t * HW * 16, (int)(T * HW * 16), pkx2,
        (t == 0) ? dbuf : dbuf + (size_t)(t - 1) * HW * 32, (int)(T * HW * 32),
        pkh2, l2b, cbuf, dbuf + (size_t)t * HW * 32, (t == 0) ? 1 : 0);
  }

  // decoder conv_transpose (stride 1 SAME == plain conv): 16 -> 64,
  // bias+lrelu+bn3, straight to d_out
  conv3x3_bn<16, 32, 64><<<convGrid, blk, 0, stream>>>(
      dbuf, pkdec, decb, bn3g, bn3b, bn3m, bn3v, (float*)d_out);
}